// Ouptut_Analyzer_51943334478455
// MI455X (gfx1250) — compile-verified
//
#include <hip/hip_runtime.h>
#include <stdint.h>

typedef __attribute__((ext_vector_type(2))) float v2f;
typedef __attribute__((ext_vector_type(8))) float v8f;
typedef unsigned long long u64;

#define BATCH  1024
#define SLOTS  13
#define FEAT   18
#define DIM    234          // SLOTS*FEAT
#define DPAD   240          // padded K (multiple of 4)
#define NWORDS 37610
#define NPHON  42
#define NTILES ((NWORDS + 15) / 16)   // 2351
#define MTILES (BATCH / 16)           // 64

// ---- workspace layout (bytes) ----
constexpr size_t OP_OFF   = 0;
constexpr size_t LOGO_OFF = OP_OFF   + (size_t)BATCH * DPAD * 4;
constexpr size_t L1M_OFF  = LOGO_OFF + (size_t)BATCH * DPAD * 4;
constexpr size_t OSQ_OFF  = L1M_OFF  + (size_t)BATCH * DPAD * 4;
constexpr size_t SLOG_OFF = OSQ_OFF  + (size_t)BATCH * 4;
constexpr size_t LSQ_OFF  = SLOG_OFF + (size_t)BATCH * 4;
constexpr size_t KCS_OFF  = (LSQ_OFF + (size_t)NWORDS * 4 + 7) & ~(size_t)7;
constexpr size_t KMSE_OFF = KCS_OFF  + (size_t)BATCH * 8;
constexpr size_t KCE_OFF  = KMSE_OFF + (size_t)BATCH * 8;
constexpr size_t PRON_OFF = KCE_OFF  + (size_t)BATCH * 8;

__device__ __forceinline__ unsigned f2ord(float f) {
    unsigned u = __float_as_uint(f);
    return (u & 0x80000000u) ? ~u : (u | 0x80000000u);
}

// ---------- stage 0: init reduction slots ----------
__global__ void k_init(u64* kcs, u64* kmse, u64* kce, int* pron) {
    int b = blockIdx.x * blockDim.x + threadIdx.x;
    if (b < BATCH) {
        kcs[b]  = 0ull;
        kmse[b] = ~0ull;
        kce[b]  = ~0ull;
        pron[b] = 1;
    }
}

// ---------- stage 1: padded row matrices + per-row stats ----------
__global__ void k_prep_rows(const float* __restrict__ in,
                            float* __restrict__ OP, float* __restrict__ LOGO,
                            float* __restrict__ L1M, float* __restrict__ OSQ,
                            float* __restrict__ SLOG) {
    const int b = blockIdx.x;
    const int j = threadIdx.x;
    __shared__ float s1[256];
    __shared__ float s2[256];
    float xsq = 0.f, sl = 0.f;
    if (j < DPAD) {
        float x = 0.f, lo = 0.f, l1 = 0.f;
        if (j < DIM) {
            x  = in[(size_t)b * DIM + j];
            lo = logf(x + 1e-8f);
            l1 = logf(1.0f - x + 1e-8f);
            xsq = x * x;
            sl  = l1;
        }
        OP[(size_t)b * DPAD + j]   = x;
        LOGO[(size_t)b * DPAD + j] = lo;
        L1M[(size_t)b * DPAD + j]  = l1;
    }
    s1[j] = xsq; s2[j] = sl;
    __syncthreads();
    for (int s = 128; s > 0; s >>= 1) {
        if (j < s) { s1[j] += s1[j + s]; s2[j] += s2[j + s]; }
        __syncthreads();
    }
    if (j == 0) { OSQ[b] = s1[0]; SLOG[b] = s2[0]; }
}

// ---------- stage 1b: per-label squared norms ----------
__global__ void k_prep_labels(const float* __restrict__ L, float* __restrict__ LSQ) {
    int n = blockIdx.x * blockDim.x + threadIdx.x;
    if (n >= NWORDS) return;
    const float* p = L + (size_t)n * DIM;
    float s = 0.f;
    for (int j = 0; j < DIM; ++j) { float v = p[j]; s += v * v; }
    LSQ[n] = s;
}

// ---------- phoneme pass ----------
__global__ void k_phoneme(const float* __restrict__ in, const float* __restrict__ plab,
                          const int* __restrict__ pli, int* __restrict__ pron,
                          float* __restrict__ out) {
    __shared__ float psq[NPHON];
    const int t = threadIdx.x;
    if (t < NPHON) {
        float s = 0.f;
        for (int j = 0; j < FEAT; ++j) { float v = plab[t * FEAT + j]; s += v * v; }
        psq[t] = s;
    }
    __syncthreads();
    const int i = blockIdx.x * blockDim.x + t;
    if (i >= BATCH * SLOTS) return;
    float x[FEAT];
    for (int j = 0; j < FEAT; ++j) x[j] = in[(size_t)i * FEAT + j];
    float best = -3.0e38f;
    int bi = 0;
    for (int p = 0; p < NPHON; ++p) {
        float d = 0.f;
        for (int j = 0; j < FEAT; ++j) d += x[j] * plab[p * FEAT + j];
        float cs = d * rsqrtf(psq[p]);   // s_norm > 0 constant per slot: argmax invariant
        if (cs > best) { best = cs; bi = p; }
    }
    out[(size_t)9 * BATCH + i] = (float)bi;           // pattern_argmax
    if (bi != pli[i]) atomicAnd(&pron[i / SLOTS], 0);
}

// ---------- main GEMM over the word-label bank (fp32 WMMA) ----------
__global__ void __launch_bounds__(256) k_gemm(
    const float* __restrict__ OP, const float* __restrict__ LOGO,
    const float* __restrict__ L1M, const float* __restrict__ Lb,
    const float* __restrict__ OSQ, const float* __restrict__ SLOG,
    const float* __restrict__ LSQ, const int* __restrict__ WLI,
    u64* __restrict__ kcs, u64* __restrict__ kmse, u64* __restrict__ kce,
    float* __restrict__ out) {
    __shared__ float Lt[16 * DPAD];
    __shared__ float lsq_s[16];
    const int tid = threadIdx.x;
    const int nbase = blockIdx.x * 16;

    // stage 16 labels x 240 (zero-padded) into LDS
    for (int i = tid; i < 16 * DPAD; i += 256) {
        int r = i / DPAD, c = i - r * DPAD;
        int n = nbase + r;
        float v = 0.f;
        if (n < NWORDS && c < DIM) v = Lb[(size_t)n * DIM + c];
        Lt[i] = v;
    }
    if (tid < 16) {
        int n = nbase + tid;
        lsq_s[tid] = (n < NWORDS) ? LSQ[n] : 0.f;
    }
    __syncthreads();

    const int wave = tid >> 5, lane = tid & 31;
    const int half = lane >> 4;        // K/M sub-group select
    const int l15  = lane & 15;
    const int koff = half << 1;        // A/B frag: half-wave holds K+2,K+3
    const int ncol = nbase + l15;
    const bool nvalid = (ncol < NWORDS);
    const float lsq   = lsq_s[l15];
    const float lnorm = sqrtf(lsq);

    for (int mt = wave; mt < MTILES; mt += 8) {
        const int arow = mt * 16 + l15;
        const float* pO = OP   + (size_t)arow * DPAD + koff;
        const float* pG = LOGO + (size_t)arow * DPAD + koff;
        const float* pH = L1M  + (size_t)arow * DPAD + koff;
        const float* pB = Lt + (size_t)l15 * DPAD + koff;
        v8f accD = {}; v8f accG = {}; v8f accH = {};
        #pragma unroll 4
        for (int t = 0; t < DPAD / 4; ++t) {
            v2f a0 = *(const v2f*)(pO + 4 * t);
            v2f a1 = *(const v2f*)(pG + 4 * t);
            v2f a2 = *(const v2f*)(pH + 4 * t);
            v2f bb = *(const v2f*)(pB + 4 * t);
            accD = __builtin_amdgcn_wmma_f32_16x16x4_f32(false, a0, false, bb, (short)0, accD, false, false);
            accG = __builtin_amdgcn_wmma_f32_16x16x4_f32(false, a1, false, bb, (short)0, accG, false, false);
            accH = __builtin_amdgcn_wmma_f32_16x16x4_f32(false, a2, false, bb, (short)0, accH, false, false);
        }
        // epilogue: C/D layout -> lane l15 = column, VGPR v = row v (+8 for upper half)
        #pragma unroll
        for (int v = 0; v < 8; ++v) {
            const int brow = mt * 16 + v + (half << 3);
            const float osq   = OSQ[brow];
            const float onorm = sqrtf(osq);
            const float slog  = SLOG[brow];
            const float dot = accD[v], lo = accG[v], l1 = accH[v];
            const float cs  = dot / (lnorm * onorm);
            const float sq  = fmaxf(osq + lsq - 2.0f * dot, 0.0f);
            const float mse = sq * (1.0f / DIM);
            const float ed  = sqrtf(sq);
            const float ce  = -(lo + slog - l1) * (1.0f / DIM);
            if (nvalid && WLI[brow] == ncol) {        // gather at word_label_indices
                out[brow]             = cs;
                out[BATCH + brow]     = mse;
                out[2 * BATCH + brow] = ed;
                out[3 * BATCH + brow] = ce;
            }
            // (ordered_value << 32 | index) keys; jnp tie-break = lowest index
            u64 ka = nvalid ? (((u64)f2ord(cs)  << 32) | (u64)(0xFFFFFFFFu - (unsigned)ncol)) : 0ull;
            u64 kb = nvalid ? (((u64)f2ord(mse) << 32) | (u64)(unsigned)ncol) : ~0ull;
            u64 kc = nvalid ? (((u64)f2ord(ce)  << 32) | (u64)(unsigned)ncol) : ~0ull;
            #pragma unroll
            for (int off = 1; off < 16; off <<= 1) {
                u64 oa = __shfl_xor(ka, off);
                u64 ob = __shfl_xor(kb, off);
                u64 oc = __shfl_xor(kc, off);
                ka = (oa > ka) ? oa : ka;
                kb = (ob < kb) ? ob : kb;
                kc = (oc < kc) ? oc : kc;
            }
            if (l15 == 0) {
                atomicMax(&kcs[brow], ka);
                atomicMin(&kmse[brow], kb);
                atomicMin(&kce[brow], kc);
            }
        }
    }
}

// ---------- finalize accuracy flags ----------
__global__ void k_final(const u64* __restrict__ kcs, const u64* __restrict__ kmse,
                        const u64* __restrict__ kce, const int* __restrict__ pron,
                        const int* __restrict__ WLI, float* __restrict__ out) {
    int b = blockIdx.x * blockDim.x + threadIdx.x;
    if (b >= BATCH) return;
    const int idx = WLI[b];
    const int ncs  = (int)(0xFFFFFFFFu - (unsigned)(kcs[b] & 0xFFFFFFFFull));
    const int nmse = (int)(unsigned)(kmse[b] & 0xFFFFFFFFull);
    const int nce  = (int)(unsigned)(kce[b] & 0xFFFFFFFFull);
    out[4 * BATCH + b] = (idx == ncs)  ? 1.f : 0.f;   // acc_max_cs
    out[5 * BATCH + b] = (idx == nmse) ? 1.f : 0.f;   // acc_min_mse
    out[6 * BATCH + b] = (idx == nmse) ? 1.f : 0.f;   // acc_min_ed (sqrt monotone)
    out[7 * BATCH + b] = (idx == nce)  ? 1.f : 0.f;   // acc_min_ce
    out[8 * BATCH + b] = pron[b] ? 1.f : 0.f;         // acc_pron
}

extern "C" void kernel_launch(void* const* d_in, const int* in_sizes, int n_in,
                              void* d_out, int out_size, void* d_ws, size_t ws_size,
                              hipStream_t stream) {
    const float* inp = (const float*)d_in[0];   // [B,S,F]
    const int*   wli = (const int*)d_in[1];     // [B]
    const int*   pli = (const int*)d_in[2];     // [B,S]
    const float* awl = (const float*)d_in[3];   // [N,S,F]
    const float* apl = (const float*)d_in[4];   // [P,F]
    float* out = (float*)d_out;

    char* ws = (char*)d_ws;
    float* OP   = (float*)(ws + OP_OFF);
    float* LOGO = (float*)(ws + LOGO_OFF);
    float* L1M  = (float*)(ws + L1M_OFF);
    float* OSQ  = (float*)(ws + OSQ_OFF);
    float* SLOG = (float*)(ws + SLOG_OFF);
    float* LSQ  = (float*)(ws + LSQ_OFF);
    u64*   KCS  = (u64*)(ws + KCS_OFF);
    u64*   KMSE = (u64*)(ws + KMSE_OFF);
    u64*   KCE  = (u64*)(ws + KCE_OFF);
    int*   PRON = (int*)(ws + PRON_OFF);

    k_init<<<(BATCH + 255) / 256, 256, 0, stream>>>(KCS, KMSE, KCE, PRON);
    k_prep_rows<<<BATCH, 256, 0, stream>>>(inp, OP, LOGO, L1M, OSQ, SLOG);
    k_prep_labels<<<(NWORDS + 255) / 256, 256, 0, stream>>>(awl, LSQ);
    k_phoneme<<<(BATCH * SLOTS + 255) / 256, 256, 0, stream>>>(inp, apl, pli, PRON, out);
    k_gemm<<<NTILES, 256, 0, stream>>>(OP, LOGO, L1M, awl, OSQ, SLOG, LSQ, wli,
                                       KCS, KMSE, KCE, out);
    k_final<<<(BATCH + 255) / 256, 256, 0, stream>>>(KCS, KMSE, KCE, PRON, wli, out);
}